// QuantumSummation_19258633355996
// MI455X (gfx1250) — compile-verified
//
#include <hip/hip_runtime.h>
#include <hip/hip_bf16.h>

// Mean of cos(w * phi) over 2^25 f32 elements.
// Memory-bound streaming reduction: b128 loads, per-thread f32 accumulation,
// WMMA (v_wmma_f32_16x16x4_f32) intra-wave reduction, deterministic two-pass.

typedef __attribute__((ext_vector_type(2))) float v2f;
typedef __attribute__((ext_vector_type(8))) float v8f;

// Exact fixed-order wave32 sum using V_WMMA_F32_16X16X4_F32.
// A (16x4 f32): lane L (0-15) VGPR0 = A[L][0]; lane L (16-31) VGPR0 = A[L-16][2].
// We place acc in A-vgpr0 and 0 in A-vgpr1, B = ones(4x16) (layout irrelevant),
// C = 0  =>  D[m][n] = acc[m] + acc[m+16] for every column n.
// Lane n holds D[0..7][n] (n<16) or D[8..15][n-16] (n>=16) in its 8 D VGPRs;
// summing those 8 and xor-shuffling across the half-wave yields the full sum
// in every lane. EXEC must be all-ones at the call site (full block, no
// divergence).
__device__ __forceinline__ float wave_reduce_wmma(float acc) {
    v2f a; a.x = acc;  a.y = 0.0f;
    v2f b; b.x = 1.0f; b.y = 1.0f;
    v8f c = {};
    v8f d = __builtin_amdgcn_wmma_f32_16x16x4_f32(
        /*neg_a=*/false, a, /*neg_b=*/false, b,
        /*c_mod=*/(short)0, c, /*reuse_a=*/false, /*reuse_b=*/false);
    float t = ((d[0] + d[1]) + (d[2] + d[3])) + ((d[4] + d[5]) + (d[6] + d[7]));
    t += __shfl_xor(t, 16, 32);
    return t;  // full 32-lane sum, broadcast to all lanes
}

__global__ __launch_bounds__(256)
void qs_partial_kernel(const float* __restrict__ phi,
                       const float* __restrict__ wts,
                       float* __restrict__ partials,
                       int n_elems) {
    const int tid   = threadIdx.x;
    const int gid   = blockIdx.x * blockDim.x + tid;
    const int gsz   = gridDim.x * blockDim.x;
    const int nvec  = n_elems >> 2;   // float4 count
    const int rem   = n_elems & 3;

    const float4* __restrict__ phi4 = reinterpret_cast<const float4*>(phi);
    const float4* __restrict__ wts4 = reinterpret_cast<const float4*>(wts);

    float acc = 0.0f;
    for (int i = gid; i < nvec; i += gsz) {
        float4 p = phi4[i];
        float4 w = wts4[i];
        acc += __cosf(w.x * p.x);
        acc += __cosf(w.y * p.y);
        acc += __cosf(w.z * p.z);
        acc += __cosf(w.w * p.w);
    }
    // Defensive tail (empty for 2^25); folded in before any reduction.
    if (gid == 0) {
        for (int j = n_elems - rem; j < n_elems; ++j)
            acc += __cosf(wts[j] * phi[j]);
    }

    // All 256 threads converged here: EXEC all-ones for WMMA.
    float wsum = wave_reduce_wmma(acc);

    __shared__ float sdata[8];
    const int wave = tid >> 5;
    const int lane = tid & 31;
    if (lane == 0) sdata[wave] = wsum;
    __syncthreads();

    if (tid == 0) {
        float s = 0.0f;
        const int nwaves = blockDim.x >> 5;
        for (int i = 0; i < nwaves; ++i) s += sdata[i];  // fixed order
        partials[blockIdx.x] = s;
    }
}

__global__ __launch_bounds__(256)
void qs_final_kernel(const float* __restrict__ partials,
                     int n_partials,
                     float addN, float invN,
                     float* __restrict__ out) {
    const int tid = threadIdx.x;

    float acc = 0.0f;
    for (int i = tid; i < n_partials; i += 256)  // fixed order per thread
        acc += partials[i];

    float wsum = wave_reduce_wmma(acc);

    __shared__ float sdata[8];
    const int wave = tid >> 5;
    const int lane = tid & 31;
    if (lane == 0) sdata[wave] = wsum;
    __syncthreads();

    if (tid == 0) {
        float s = 0.0f;
        for (int i = 0; i < 8; ++i) s += sdata[i];
        out[0] = (s + addN) * invN;
    }
}

extern "C" void kernel_launch(void* const* d_in, const int* in_sizes, int n_in,
                              void* d_out, int out_size, void* d_ws, size_t ws_size,
                              hipStream_t stream) {
    const float* phi = reinterpret_cast<const float*>(d_in[0]);
    const float* wts = reinterpret_cast<const float*>(d_in[1]);
    float* out      = reinterpret_cast<float*>(d_out);
    float* partials = reinterpret_cast<float*>(d_ws);

    const int C = in_sizes[0];

    // N = next power of two >= C (reference: 2^ceil(log2(C)))
    long long N = 1;
    while (N < (long long)C) N <<= 1;
    const float addN = (float)(N - (long long)C);
    const float invN = 1.0f / (float)N;

    // 2048 blocks x 256 threads: 524288 threads, 16 float4 loads per thread
    // per input at C = 2^25. Clamp to workspace capacity defensively.
    int nblocks = 2048;
    const int max_partials = (int)(ws_size / sizeof(float));
    if (nblocks > max_partials) nblocks = max_partials > 0 ? max_partials : 1;

    qs_partial_kernel<<<nblocks, 256, 0, stream>>>(phi, wts, partials, C);
    qs_final_kernel<<<1, 256, 0, stream>>>(partials, nblocks, addN, invN, out);
}